// GaussianSplatting_12000138625130
// MI455X (gfx1250) — compile-verified
//
#include <hip/hip_runtime.h>
#include <stdint.h>

// ---- CDNA5 vector types ----
typedef float v2f  __attribute__((ext_vector_type(2)));
typedef float v8f  __attribute__((ext_vector_type(8)));
typedef unsigned int u32x4 __attribute__((ext_vector_type(4)));
typedef int   i32x8 __attribute__((ext_vector_type(8)));
typedef int   i32x4 __attribute__((ext_vector_type(4)));

#define TPB   256
#define PTS   240           // points per block (8 waves * 30 pts; 6 WMMAs/wave * 5 pts)
#define FPB   (PTS * 9)     // floats per R/S/M/cov tile = 2160
#define SLICE 270           // FPB / 8 waves

// One TDM descriptor load: copies nFloats floats from global memory at gsrc to the LDS byte address lds_byte
// Encodes a 1-row 2D tensor (tile_dim0 = nFloats; 0 => hardware NOP).
__device__ __forceinline__ void tdm_load_to_lds(unsigned lds_byte,
                                                const float* gsrc, int nFloats)
{
    if (nFloats < 0) nFloats = 0;                 // clamp; tile_dim0==0 => NOP
    uint64_t ga = (uint64_t)(uintptr_t)gsrc;
    u32x4 g0;
    g0[0] = 1u;                                   // count=1, user descriptor
    g0[1] = lds_byte;                             // lds_addr (bytes)
    g0[2] = (unsigned)ga;                         // global_addr[31:0]
    g0[3] = ((unsigned)(ga >> 32) & 0x01FFFFFFu)  // global_addr[56:32]
            | 0x80000000u;                        // type = 2 ("image")
    i32x8 g1;
    g1[0] = 0x00020000;                           // data_size = 2 (4 bytes)
    g1[1] = (nFloats & 0xFFFF) << 16;             // tensor_dim0[15:0]
    g1[2] = (int)(((unsigned)nFloats >> 16) | (1u << 16)); // dim0[31:16], tensor_dim1=1
    g1[3] = nFloats << 16;                        // tile_dim0
    g1[4] = 1;                                    // tile_dim1=1, tile_dim2=0
    g1[5] = nFloats;                              // tensor_dim0_stride[31:0]
    g1[6] = 0;
    g1[7] = 0;
    i32x4 z4 = {0, 0, 0, 0};
    i32x8 z8 = {0, 0, 0, 0, 0, 0, 0, 0};
    __builtin_amdgcn_tensor_load_to_lds(g0, g1, z4, z4, z8, 0);
}

__global__ __launch_bounds__(TPB)
void gsplat_kernel(const float* __restrict__ pos, const float* __restrict__ col,
                   const float* __restrict__ alp, const float* __restrict__ Rm,
                   const float* __restrict__ Sm,  float* __restrict__ out, int N)
{
    __shared__ float Rt[FPB], St[FPB], Mb[FPB], Cb[FPB];

    const int tid = threadIdx.x;
    const int wv   = tid >> 5;          // wave 0..7 (wave32)
    const int lane = tid & 31;
    const int blockBase = blockIdx.x * PTS;
    const int nPts   = min(PTS, N - blockBase);
    const int totalF = nPts * 9;

    float* out_pos = out;
    float* out_cov = out + (size_t)3  * N;
    float* out_alp = out + (size_t)12 * N;
    float* out_sh  = out + (size_t)13 * N;

    // ---- Phase 0: async TDM tile loads of R and S into LDS (one slice per wave)
    const int sBeg = wv * SLICE;
    const int sCnt = min(SLICE, totalF - sBeg);
    const unsigned ldsR = (unsigned)(uintptr_t)&Rt[0];   // low 32 bits of flat = LDS offset
    const unsigned ldsS = (unsigned)(uintptr_t)&St[0];
    tdm_load_to_lds(ldsR + (unsigned)sBeg * 4u, Rm + (size_t)blockBase * 9 + sBeg, sCnt);
    tdm_load_to_lds(ldsS + (unsigned)sBeg * 4u, Sm + (size_t)blockBase * 9 + sBeg, sCnt);

    // ---- Overlap with TDM: stream positions / alphas / SH (pure bandwidth)
    for (int i = tid; i < nPts * 3; i += TPB)
        out_pos[(size_t)blockBase * 3 + i] = pos[(size_t)blockBase * 3 + i];
    if (tid < nPts) {
        const int g = blockBase + tid;
        out_alp[g] = alp[g];
        const float c0 = col[(size_t)g * 3 + 0];
        const float c1 = col[(size_t)g * 3 + 1];
        const float c2 = col[(size_t)g * 3 + 2];
        out_sh[(size_t)g * 4 + 0] = 0.282095f;        // SH_C0
        out_sh[(size_t)g * 4 + 1] = c1 * 0.488603f;   // SH_C1 * g
        out_sh[(size_t)g * 4 + 2] = c0 * 0.488603f;   // SH_C1 * r
        out_sh[(size_t)g * 4 + 3] = c2 * 0.488603f;   // SH_C1 * b
    }

    __builtin_amdgcn_s_wait_tensorcnt(0);   // my wave's TDM slices landed
    __syncthreads();                        // everyone's slices landed

    // ---- Phase 1: M = R * S per point (27 FMAs, lane-local; stride-9 => bank-clean)
    if (tid < nPts) {
        float r_[9], s_[9];
        #pragma unroll
        for (int k = 0; k < 9; ++k) { r_[k] = Rt[tid * 9 + k]; s_[k] = St[tid * 9 + k]; }
        #pragma unroll
        for (int i = 0; i < 3; ++i)
            #pragma unroll
            for (int j = 0; j < 3; ++j)
                Mb[tid * 9 + i * 3 + j] =
                    r_[i * 3 + 0] * s_[0 + j] + r_[i * 3 + 1] * s_[3 + j] + r_[i * 3 + 2] * s_[6 + j];
    }
    __syncthreads();

    // ---- Phase 2: cov = M * M^T via V_WMMA_F32_16X16X4_F32.
    // A operand: 5 points' M as block rows (15 rows used, K=0..2 used, rest 0).
    // Same registers fed as B are interpreted as A^T (f32 A/B layouts are mutual
    // transposes), so D = A * A^T; diagonal 3x3 blocks = per-point M*M^T.
    {
        const int row  = lane & 15;     // A row / D column index n
        const int half = lane >> 4;     // 0: K=0,1 | 1: K=2,3
        const int i3   = row % 3;
        const int pp   = row / 3;       // local point within the 5-point group
        #pragma unroll
        for (int q = 0; q < 6; ++q) {
            const int  P0 = wv * 30 + q * 5;
            const int  p  = P0 + pp;
            const bool av = (row < 15) && (p < nPts);
            const int  mbase = av ? (p * 9 + i3 * 3) : 0;     // clamp to stay in-bounds
            float m0 = Mb[mbase + 0], m1 = Mb[mbase + 1], m2 = Mb[mbase + 2];
            m0 = av ? m0 : 0.0f; m1 = av ? m1 : 0.0f; m2 = av ? m2 : 0.0f;
            v2f a;
            a.x = half ? m2   : m0;
            a.y = half ? 0.0f : m1;
            v8f c = {};
            v8f d = __builtin_amdgcn_wmma_f32_16x16x4_f32(
                        false, a, false, a, (short)0, c, false, false);
            // D layout: lanes 0-15 -> (m=r, n=lane); lanes 16-31 -> (m=r+8, n=lane-16)
            #pragma unroll
            for (int r8 = 0; r8 < 8; ++r8) {
                const int m  = r8 + half * 8;
                const int pm = m / 3;
                if ((m < 15) && (row < 15) && (pm == pp) && (P0 + pm < nPts))
                    Cb[(P0 + pm) * 9 + (m % 3) * 3 + i3] = d[r8];
            }
        }
    }
    __syncthreads();

    // ---- Phase 3: dense coalesced store of the cov tile
    for (int i = tid; i < totalF; i += TPB)
        out_cov[(size_t)blockBase * 9 + i] = Cb[i];
}

extern "C" void kernel_launch(void* const* d_in, const int* in_sizes, int n_in,
                              void* d_out, int out_size, void* d_ws, size_t ws_size,
                              hipStream_t stream)
{
    (void)n_in; (void)out_size; (void)d_ws; (void)ws_size;
    const float* pos = (const float*)d_in[0];
    const float* col = (const float*)d_in[1];
    const float* alp = (const float*)d_in[2];
    const float* Rm  = (const float*)d_in[3];
    const float* Sm  = (const float*)d_in[4];
    float* out = (float*)d_out;
    const int N = in_sizes[2];                 // alphas: one per point
    const int grid = (N + PTS - 1) / PTS;
    hipLaunchKernelGGL(gsplat_kernel, dim3(grid), dim3(TPB), 0, stream,
                       pos, col, alp, Rm, Sm, out, N);
}